// ImproveFFBlock4_69715909149112
// MI455X (gfx1250) — compile-verified
//
#include <hip/hip_runtime.h>
#include <cstdint>
#include <math.h>

// ---------------------------------------------------------------------------
// Problem geometry (fixed by the reference):
//   x1: (2,16384,32,32)  pool2 -> (.,.,16,16), tiled x2
//   x2: (2, 4096,64,64)  pool4 -> tiled x8
//   x3: (2,  256,128,128) pool8 -> tiled x128
//   x4: (2,    1,256,256) pool16 -> tiled x32768
//   ff : (2,32768,16,16)
//   out = relu(y1+y2+y3+y4+ff)  (2,32768,16,16) fp32
// Pure streaming: ~437 MB HBM traffic -> ~19 us floor @ 23.3 TB/s.
// ---------------------------------------------------------------------------

namespace {
constexpr int B     = 2;
constexpr int C1    = 16384;   // x1 channels (half the output channels)
constexpr int C2    = 4096;
constexpr int C3    = 256;
constexpr int OUT_C = 2 * C1;  // 32768
constexpr int PX    = 256;     // 16*16 output pixels per channel
}

// CDNA5 async global->LDS copy path (ASYNCcnt-tracked), if the toolchain
// exposes the gfx1250 builtin.  Falls back to direct float4 loads otherwise.
#if defined(__gfx1250__)
#  if __has_builtin(__builtin_amdgcn_global_load_async_to_lds_b128)
#    define HAVE_ASYNC_LDS 1
#  endif
#endif
#ifndef HAVE_ASYNC_LDS
#  define HAVE_ASYNC_LDS 0
#endif

#if HAVE_ASYNC_LDS
typedef int v4i_async __attribute__((ext_vector_type(4)));
#endif

// ---------------------------------------------------------------------------
// Pass 1: pool the three "small-output" tensors into workspace.
// One block (256 threads) per (b,c) channel; thread t -> output pixel (h,w).
// For kernel size K (multiple of 4), each K-wide pooling row along w is K/4
// naturally-aligned float4 loads -> all traffic is GLOBAL_LOAD_B128.
// ---------------------------------------------------------------------------
template <int K, int IW>
__device__ __forceinline__ void pool_body(const float* __restrict__ in,
                                          float* __restrict__ outp, int bc) {
  const int t = (int)threadIdx.x;
  const int h = t >> 4;
  const int w = t & 15;
  const float* src = in + (size_t)bc * (size_t)(IW * IW) +
                     (size_t)(h * K) * IW + (size_t)(w * K);
  float mx = -__builtin_inff();
#pragma unroll
  for (int r = 0; r < K; ++r) {
    const float* row = src + (size_t)r * IW;
#pragma unroll
    for (int q = 0; q < K / 4; ++q) {
      const float4 v = *(const float4*)(row + q * 4);
      mx = fmaxf(mx, fmaxf(fmaxf(v.x, v.y), fmaxf(v.z, v.w)));
    }
  }
  outp[(size_t)bc * PX + t] = mx;
}

__global__ __launch_bounds__(256) void pool_all_kernel(
    const float* __restrict__ x2, const float* __restrict__ x3,
    const float* __restrict__ x4, float* __restrict__ p2,
    float* __restrict__ p3, float* __restrict__ p4) {
  const int blk = (int)blockIdx.x;
  if (blk < B * C2) {
    pool_body<4, 64>(x2, p2, blk);                       // 128 MiB stream
  } else if (blk < B * C2 + B * C3) {
    pool_body<8, 128>(x3, p3, blk - B * C2);             // 32 MiB stream
  } else {
    pool_body<16, 256>(x4, p4, blk - (B * C2 + B * C3)); // 0.5 MiB
  }
}

// ---------------------------------------------------------------------------
// Pass 2: fused 2x2 pool of x1 + broadcast sums + ReLU + both tile copies.
// One block = 4 consecutive (b,c) channels of x1 (4 KiB each).
//   thread t: sub-channel ch = t>>6; within channel: h = (t&63)>>2,
//   4 output pixels along w starting at 4*(t&3).
// x1 is read exactly once; the pooled value s is reused for out[c] and
// out[c+16384].  P2/P3/P4 reads are L2 hits (just produced, <= 8.5 MiB).
// ---------------------------------------------------------------------------
__global__ __launch_bounds__(256) void fused_kernel(
    const float* __restrict__ x1, const float* __restrict__ ff,
    const float* __restrict__ p2, const float* __restrict__ p3,
    const float* __restrict__ p4, float* __restrict__ out) {
  const int t   = (int)threadIdx.x;
  const int blk = (int)blockIdx.x;   // 0 .. B*C1/4 - 1
  const int ch  = t >> 6;            // 0..3 sub-channel within block
  const int tt  = t & 63;
  const int bc  = blk * 4 + ch;      // flat (b,c) over x1 channels
  const int b   = bc >> 14;          // / C1
  const int c   = bc & (C1 - 1);
  const int h   = tt >> 2;           // 0..15 output row
  const int m   = tt & 3;            // float4 group along w
  const int px  = h * 16 + m * 4;

  float4 a0, a1, r0v, r1v;

#if HAVE_ASYNC_LDS
  // Stage the block's 16 KiB x1 slab into LDS with async B128 copies
  // (coalesced 16B chunks, 4 per thread), padded rows to spread LDS banks.
  constexpr int ROWF = 36;            // padded row stride in floats (144 B)
  constexpr int CHF  = 32 * ROWF;     // 1152 floats per staged channel
  __shared__ float smem[4 * CHF];
  {
    const float* slab = x1 + (size_t)blk * 4096;  // 4 ch * 1024 floats
#pragma unroll
    for (int it = 0; it < 4; ++it) {
      const int i   = t + it * 256;   // 16-byte chunk id, 0..1023
      const int cc  = i >> 8;         // channel within slab
      const int ci  = i & 255;        // chunk within channel
      const int r   = ci >> 3;        // input row (8 chunks per 32-fl row)
      const int q   = ci & 7;
      const int off = cc * CHF + r * ROWF + q * 4;
      __builtin_amdgcn_global_load_async_to_lds_b128(
          (v4i_async*)(slab + (size_t)i * 4),
          (v4i_async*)(&smem[off]),
          0, 0);
    }
  }
#if __has_builtin(__builtin_amdgcn_s_wait_asynccnt)
  __builtin_amdgcn_s_wait_asynccnt(0);
#else
  asm volatile("s_wait_asynccnt 0" ::: "memory");
#endif
  __syncthreads();
  const float* cb = &smem[ch * CHF];
  a0  = *(const float4*)(cb + (2 * h) * ROWF + 8 * m);
  a1  = *(const float4*)(cb + (2 * h) * ROWF + 8 * m + 4);
  r0v = *(const float4*)(cb + (2 * h + 1) * ROWF + 8 * m);
  r1v = *(const float4*)(cb + (2 * h + 1) * ROWF + 8 * m + 4);
#else
  const float* xc = x1 + (size_t)bc * 1024;
  a0  = *(const float4*)(xc + (2 * h) * 32 + 8 * m);
  a1  = *(const float4*)(xc + (2 * h) * 32 + 8 * m + 4);
  r0v = *(const float4*)(xc + (2 * h + 1) * 32 + 8 * m);
  r1v = *(const float4*)(xc + (2 * h + 1) * 32 + 8 * m + 4);
#endif

  // 2x2 max pool of x1 -> 4 output pixels
  float4 s;
  s.x = fmaxf(fmaxf(a0.x, a0.y), fmaxf(r0v.x, r0v.y));
  s.y = fmaxf(fmaxf(a0.z, a0.w), fmaxf(r0v.z, r0v.w));
  s.z = fmaxf(fmaxf(a1.x, a1.y), fmaxf(r1v.x, r1v.y));
  s.w = fmaxf(fmaxf(a1.z, a1.w), fmaxf(r1v.z, r1v.w));

  // Broadcast terms (L2-resident)
  const float4 v2 = *(const float4*)(p2 + (size_t)(b * C2 + (c & (C2 - 1))) * PX + px);
  const float4 v3 = *(const float4*)(p3 + (size_t)(b * C3 + (c & (C3 - 1))) * PX + px);
  const float4 v4 = *(const float4*)(p4 + (size_t)b * PX + px);
  s.x += v2.x + v3.x + v4.x;
  s.y += v2.y + v3.y + v4.y;
  s.z += v2.z + v3.z + v4.z;
  s.w += v2.w + v3.w + v4.w;

  // Both tile copies share s; only ff differs.
  const size_t o0 = ((size_t)b * OUT_C + (size_t)c) * PX + (size_t)px;
  const size_t o1 = o0 + (size_t)C1 * PX;
  const float4 f0 = *(const float4*)(ff + o0);
  const float4 f1 = *(const float4*)(ff + o1);
  float4 r0, r1;
  r0.x = fmaxf(s.x + f0.x, 0.0f);
  r0.y = fmaxf(s.y + f0.y, 0.0f);
  r0.z = fmaxf(s.z + f0.z, 0.0f);
  r0.w = fmaxf(s.w + f0.w, 0.0f);
  r1.x = fmaxf(s.x + f1.x, 0.0f);
  r1.y = fmaxf(s.y + f1.y, 0.0f);
  r1.z = fmaxf(s.z + f1.z, 0.0f);
  r1.w = fmaxf(s.w + f1.w, 0.0f);
  *(float4*)(out + o0) = r0;
  *(float4*)(out + o1) = r1;
}

// ---------------------------------------------------------------------------
extern "C" void kernel_launch(void* const* d_in, const int* in_sizes, int n_in,
                              void* d_out, int out_size, void* d_ws, size_t ws_size,
                              hipStream_t stream) {
  (void)in_sizes; (void)n_in; (void)out_size; (void)ws_size;

  const float* x1 = (const float*)d_in[0];
  const float* x2 = (const float*)d_in[1];
  const float* x3 = (const float*)d_in[2];
  const float* x4 = (const float*)d_in[3];
  const float* ff = (const float*)d_in[4];
  float* out = (float*)d_out;

  // Workspace layout (floats): P2 | P3 | P4   (~8.5 MiB total)
  float* p2 = (float*)d_ws;
  float* p3 = p2 + (size_t)B * C2 * PX;
  float* p4 = p3 + (size_t)B * C3 * PX;

  const int pool_blocks = B * C2 + B * C3 + B;  // 8706
  pool_all_kernel<<<pool_blocks, 256, 0, stream>>>(x2, x3, x4, p2, p3, p4);

  const int fused_blocks = (B * C1) / 4;        // 8192
  fused_kernel<<<fused_blocks, 256, 0, stream>>>(x1, ff, p2, p3, p4, out);
}